// VBGE_60593398612496
// MI455X (gfx1250) — compile-verified
//
#include <hip/hip_runtime.h>

// ---------------------------------------------------------------------------
// MI455X (gfx1250) implementation of the 2-layer DGCN reference.
// Dense GEMMs use V_WMMA_F32_16X16X4_F32 (exact fp32 matrix cores).
// SpMM (segment_sum over COO edges) uses wave-per-edge float4 gathers +
// hardware global_atomic_add_f32; gather working sets (<=31 MB) are
// L2-resident (192 MB L2), so SpMM runs near L2 bandwidth.
// ---------------------------------------------------------------------------

typedef __attribute__((ext_vector_type(2))) float v2f;
typedef __attribute__((ext_vector_type(8))) float v8f;

// W staged in LDS pair-interleaved: lds[q*256 + col*2 + s] = W[2q+s][col]
// so a B fragment {W[2q][col], W[2q+1][col]} is ONE aligned ds_load_b64.
// Footprint: 64 q-rows * 128 cols * 2 floats = 64 KB.

// C[M,128] = act( A1[M,128] @ W1 (+ A2[M,128] @ W2) + bias ), arbitrary ldc.
// 4 waves/block, 16 rows per wave, 64 rows per block.
// A-frag (16x4 f32): lane<16 -> {A[m][4k],A[m][4k+1]}, lane>=16 -> {A[m][4k+2],A[m][4k+3]}
// B-frag (4x16 f32): lane<16 -> rows {4k,4k+1}, lane>=16 -> rows {4k+2,4k+3}, col = lane&15
// C/D (16x16 f32): VGPR r -> row m0+r (lanes 0-15) / m0+8+r (lanes 16-31), col = lane&15
__global__ __launch_bounds__(128)
void gemm128_wmma(const float* __restrict__ A1, int lda1,
                  const float* __restrict__ W1,
                  const float* __restrict__ A2, int lda2,
                  const float* __restrict__ W2,
                  const float* __restrict__ bias,
                  float* __restrict__ C, int ldc,
                  int M, int act)
{
    __shared__ float lds[64 * 256];   // pair-interleaved W tile, 64 KB

    const int tid  = threadIdx.x;
    const int lane = tid & 31;
    const int wave = tid >> 5;
    const int half = lane >> 4;          // 0: lanes 0-15, 1: lanes 16-31
    const int l15  = lane & 15;
    const int m0   = blockIdx.x * 64 + wave * 16;

    v8f zero = {};
    v8f acc[8];
#pragma unroll
    for (int i = 0; i < 8; ++i) acc[i] = zero;

    // wave-uniform scalar guard -> EXEC stays all-ones around WMMA
    const int rowOK = (__builtin_amdgcn_readfirstlane(m0) < M);
    const int npass = (A2 != nullptr) ? 2 : 1;

    for (int p = 0; p < npass; ++p) {
        const float* __restrict__ Ap  = (p == 0) ? A1 : A2;
        const float* __restrict__ Wp  = (p == 0) ? W1 : W2;
        const int                 lda = (p == 0) ? lda1 : lda2;

        __syncthreads();   // previous pass finished reading LDS
        // stage W (128x128 f32) into LDS pair-interleaved, float4 per thread
        for (int i = tid; i < 4096; i += 128) {
            const float4 w = ((const float4*)Wp)[i];
            const int r = i >> 5;            // W row (32 float4 per row)
            const int c = (i & 31) << 2;     // W col
            float* d = &lds[(r >> 1) * 256 + c * 2 + (r & 1)];
            d[0] = w.x; d[2] = w.y; d[4] = w.z; d[6] = w.w;
        }
        __syncthreads();

        if (rowOK) {
            const float* __restrict__ arow =
                Ap + (size_t)(m0 + l15) * lda + 2 * half;
#pragma unroll 4
            for (int kk = 0; kk < 32; ++kk) {
                const v2f a = *(const v2f*)(arow + kk * 4);      // global_load_b64
                const float* __restrict__ bbase =
                    &lds[(kk * 2 + half) * 256 + l15 * 2];
#pragma unroll
                for (int nt = 0; nt < 8; ++nt) {
                    const v2f b = *(const v2f*)(bbase + nt * 32); // ds_load_b64
                    acc[nt] = __builtin_amdgcn_wmma_f32_16x16x4_f32(
                        false, a, false, b, (short)0, acc[nt], false, false);
                }
            }
        }
    }

    if (rowOK) {
#pragma unroll
        for (int nt = 0; nt < 8; ++nt) {
            const int col = nt * 16 + l15;
            const float bv = bias ? bias[col] : 0.0f;
#pragma unroll
            for (int r = 0; r < 8; ++r) {
                const int row = m0 + r + 8 * half;
                float v = acc[nt][r] + bv;
                if (act == 1) v = v > 0.0f ? v : 0.0f;   // ReLU
                C[(size_t)row * ldc + col] = v;
            }
        }
    }
}

// guaranteed single-instruction f32 atomic add (no CAS fallback)
__device__ __forceinline__ void atomic_add_f32_hw(float* p, float v)
{
    asm volatile("global_atomic_add_f32 %0, %1, off"
                 :
                 : "v"(p), "v"(v)
                 : "memory");
}

// out[rows[e], :] += vals[e] * X[cols[e], :]   (128 features)
// one wave per edge; each lane handles 4 consecutive features (float4 gather)
__global__ __launch_bounds__(256)
void spmm_edge_atomic(const int* __restrict__ rows, const int* __restrict__ cols,
                      const float* __restrict__ vals, const float* __restrict__ X,
                      float* __restrict__ out, int E)
{
    const int gt   = blockIdx.x * 256 + threadIdx.x;
    const int e    = gt >> 5;
    const int lane = gt & 31;
    if (e >= E) return;
    const int   r = rows[e];      // wave-uniform -> scalar loads
    const int   c = cols[e];
    const float v = vals[e];
    const float4 x = ((const float4*)(X + (size_t)c * 128))[lane];
    float* o = out + (size_t)r * 128 + lane * 4;
    atomic_add_f32_hw(o + 0, v * x.x);
    atomic_add_f32_hw(o + 1, v * x.y);
    atomic_add_f32_hw(o + 2, v * x.z);
    atomic_add_f32_hw(o + 3, v * x.w);
}

__global__ __launch_bounds__(256)
void fill_zero(float* __restrict__ p, long long n)
{
    const long long i = (long long)blockIdx.x * 256 + threadIdx.x;
    if (i < n) p[i] = 0.0f;
}

// Z = LeakyReLU(Z + bias[col], 0.2) over [n_rows, 128] row-major
__global__ __launch_bounds__(256)
void bias_leaky(float* __restrict__ Z, const float* __restrict__ b, long long total)
{
    const long long i = (long long)blockIdx.x * 256 + threadIdx.x;
    if (i < total) {
        const float v = Z[i] + b[(int)(i & 127)];
        Z[i] = v >= 0.0f ? v : 0.2f * v;
    }
}

extern "C" void kernel_launch(void* const* d_in, const int* in_sizes, int n_in,
                              void* d_out, int out_size, void* d_ws, size_t ws_size,
                              hipStream_t stream)
{
    const float* ufea   = (const float*)d_in[0];
    const float* vfea   = (const float*)d_in[1];
    const int*   uv_src = (const int*)d_in[2];
    const int*   uv_dst = (const int*)d_in[3];
    const float* uv_w   = (const float*)d_in[4];
    const float* vu_w   = (const float*)d_in[5];
    const float* gcW    = (const float*)d_in[6];
    const float* gcb    = (const float*)d_in[7];
    const float* unW    = (const float*)d_in[8];
    const float* unb    = (const float*)d_in[9];

    const int F  = 128;
    const int NU = in_sizes[0] / F;   // 60000
    const int NI = in_sizes[1] / F;   // 40000
    const int E  = in_sizes[2];       // 1e6

    // workspace: 5 buffers of [NU,128] f32 (~154 MB total)
    float* ws = (float*)d_ws;
    const size_t S = (size_t)NU * F;
    float* bufT  = ws + 0 * S;   // GEMM temp (user- or item-sized)
    float* bufT2 = ws + 1 * S;   // GEMM temp
    float* bufA  = ws + 2 * S;   // gc1/gc4 output (item-sized)
    float* bufB  = ws + 3 * S;   // gc2 output (user-sized)
    float* bufU  = ws + 4 * S;   // gc3 output (user-sized)

    float* outU = (float*)d_out;                     // [NU,256]
    float* outI = (float*)d_out + (size_t)NU * 256;  // [NI,256]

    const float* lu = ufea; int ldu = 128;
    const float* li = vfea; int ldv = 128;

    auto gemm = [&](float* C, int ldc, const float* A1, int lda1, const float* W1,
                    const float* A2, int lda2, const float* W2,
                    const float* bias, int M, int act) {
        gemm128_wmma<<<(M + 63) / 64, 128, 0, stream>>>(
            A1, lda1, W1, A2, lda2, W2, bias, C, ldc, M, act);
    };
    auto spmm = [&](float* out, int n_out, const int* rows, const int* cols,
                    const float* vals, const float* X, const float* bias) {
        const long long tot = (long long)n_out * F;
        fill_zero<<<(unsigned)((tot + 255) / 256), 256, 0, stream>>>(out, tot);
        spmm_edge_atomic<<<(int)(((long long)E * 32 + 255) / 256), 256, 0, stream>>>(
            rows, cols, vals, X, out, E);
        bias_leaky<<<(unsigned)((tot + 255) / 256), 256, 0, stream>>>(out, bias, tot);
    };

    for (int l = 0; l < 2; ++l) {
        const float* gW = gcW + (size_t)l * 4 * 128 * 128;
        const float* gb = gcb + (size_t)l * 4 * 128;
        const float* uW = unW + (size_t)l * 2 * 256 * 128;
        const float* ub = unb + (size_t)l * 2 * 128;

        // gc1: A = leaky( VU @ (lu @ W0) + b0 )  -> [NI,128]
        gemm(bufT, 128, lu, ldu, gW + 0 * 128 * 128, nullptr, 0, nullptr, nullptr, NU, 0);
        spmm(bufA, NI, uv_dst, uv_src, vu_w, bufT, gb + 0 * 128);
        // gc2: B = leaky( UV @ (li @ W1) + b1 )  -> [NU,128]
        gemm(bufT2, 128, li, ldv, gW + 1 * 128 * 128, nullptr, 0, nullptr, nullptr, NI, 0);
        spmm(bufB, NU, uv_src, uv_dst, uv_w, bufT2, gb + 1 * 128);
        // gc3: U = leaky( UV @ (A @ W2) + b2 )   -> [NU,128]
        gemm(bufT, 128, bufA, 128, gW + 2 * 128 * 128, nullptr, 0, nullptr, nullptr, NI, 0);
        spmm(bufU, NU, uv_src, uv_dst, uv_w, bufT, gb + 2 * 128);
        // gc4: I = leaky( VU @ (B @ W3) + b3 )   -> [NI,128]
        gemm(bufT2, 128, bufB, 128, gW + 3 * 128 * 128, nullptr, 0, nullptr, nullptr, NU, 0);
        spmm(bufA, NI, uv_dst, uv_src, vu_w, bufT2, gb + 3 * 128);

        // union: lu' = relu( [U, lu] @ unW0 + unb0 ) written to outU col block l
        gemm(outU + l * 128, 256, bufU, 128, uW + 0 * 256 * 128,
             lu, ldu, uW + 0 * 256 * 128 + 128 * 128, ub + 0 * 128, NU, 1);
        // union: li' = relu( [I, li] @ unW1 + unb1 ) written to outI col block l
        gemm(outI + l * 128, 256, bufA, 128, uW + 1 * 256 * 128,
             li, ldv, uW + 1 * 256 * 128 + 128 * 128, ub + 1 * 128, NI, 1);

        lu = outU; ldu = 256;   // next layer reads this layer's outputs in-place
        li = outI; ldv = 256;
    }

    (void)n_in; (void)out_size; (void)ws_size;
}